// GNN_DTDG_70171175682618
// MI455X (gfx1250) — compile-verified
//
#include <hip/hip_runtime.h>
#include <hip/hip_bf16.h>

#define N_NODES 20000
#define N_EDGES 100000
#define TPB 256

typedef __attribute__((ext_vector_type(2))) float v2f;
typedef __attribute__((ext_vector_type(8))) float v8f;

// ---------------------------------------------------------------------------
// WMMA f32 16x16x4 GEMM:  C[M x NC] = A[M x K] @ B[K x NC]   (BT: B is [NC x K])
// One 16x16 output tile per wave, 4 waves per block. All dims multiples of 16/4.
// A-layout (ISA 7.12.2): lane = (khalf<<4)|m ; VGPR0=K+0, VGPR1=K+1 (khalf adds 2)
// C-layout: VGPR r -> row r (lanes 0-15) / row r+8 (lanes 16-31), col = lane&15
// ---------------------------------------------------------------------------
template <bool BT>
__global__ void gemm_wmma_f32(const float* __restrict__ A, const float* __restrict__ B,
                              float* __restrict__ C, int M, int K, int NC,
                              int lda, int ldb, int ldc) {
    const int tilesN = NC >> 4;
    const int wave   = blockIdx.x * (blockDim.x >> 5) + (threadIdx.x >> 5);
    const int trow   = wave / tilesN;
    const int tcol   = wave - trow * tilesN;
    if ((trow << 4) >= M) return;                // whole-wave uniform exit

    const int lane = threadIdx.x & 31;
    const int mloc = lane & 15;                  // M (for A) or N (for B/C) within tile
    const int kh   = (lane >> 4) << 1;           // 0 or 2: K sub-offset per lane half
    const int row0 = trow << 4, col0 = tcol << 4;

    v8f acc = {};
    const float* aBase = A + (size_t)(row0 + mloc) * lda + kh;
    for (int k = 0; k < K; k += 4) {
        v2f a, b;
        a.x = aBase[k];
        a.y = aBase[k + 1];
        if (BT) {
            const float* bp = B + (size_t)(col0 + mloc) * ldb + k + kh;
            b.x = bp[0];
            b.y = bp[1];
        } else {
            const float* bp = B + (size_t)(k + kh) * ldb + col0 + mloc;
            b.x = bp[0];
            b.y = bp[ldb];
        }
        acc = __builtin_amdgcn_wmma_f32_16x16x4_f32(false, a, false, b,
                                                    (short)0, acc, false, false);
    }

    const int   rbase = row0 + ((lane >> 4) << 3);      // +0 or +8
    float*      cp    = C + (size_t)rbase * ldc + col0 + mloc;
#pragma unroll
    for (int r = 0; r < 8; ++r) cp[(size_t)r * ldc] = acc[r];
}

// ---------------------------------------------------------------------------
// Helpers for float segment-max via monotone uint encoding
// ---------------------------------------------------------------------------
__device__ __forceinline__ unsigned f2ord(float f) {
    unsigned u = __float_as_uint(f);
    return (u & 0x80000000u) ? ~u : (u | 0x80000000u);
}
__device__ __forceinline__ float ord2f(unsigned u) {
    return (u & 0x80000000u) ? __uint_as_float(u & 0x7FFFFFFFu) : __uint_as_float(~u);
}

__global__ void zero_kernel(float* p, int n) {
    int t = blockIdx.x * blockDim.x + threadIdx.x;
    if (t < n) p[t] = 0.0f;
}

// el/er per (node, head): dot of ft[n,h,:] with al[h,:], ar[h,:]
__global__ void node_attn_kernel(const float* __restrict__ ft, const float* __restrict__ al,
                                 const float* __restrict__ ar, float* __restrict__ el,
                                 float* __restrict__ er) {
    int t = blockIdx.x * blockDim.x + threadIdx.x;
    if (t >= N_NODES * 2) return;
    int n = t >> 1, h = t & 1;
    const float* f  = ft + (size_t)n * 128 + h * 64;
    const float* a1 = al + h * 64;
    const float* a2 = ar + h * 64;
    float s1 = 0.f, s2 = 0.f;
#pragma unroll 8
    for (int d = 0; d < 64; ++d) { float v = f[d]; s1 += v * a1[d]; s2 += v * a2[d]; }
    el[t] = s1;
    er[t] = s2;
}

// ee per (edge, head)
__global__ void edge_attn_kernel(const float* __restrict__ fe, const float* __restrict__ ae,
                                 float* __restrict__ ee) {
    int t = blockIdx.x * blockDim.x + threadIdx.x;
    if (t >= N_EDGES * 2) return;
    int e = t >> 1, h = t & 1;
    const float* f = fe + (size_t)e * 128 + h * 64;
    const float* a = ae + h * 64;
    float s = 0.f;
#pragma unroll 8
    for (int d = 0; d < 64; ++d) s += f[d] * a[d];
    ee[t] = s;
}

// logits = leaky_relu(el[src]+er[dst]+ee, 0.2); segment-max into mU[dst]
__global__ void logits_max_kernel(const float* __restrict__ el, const float* __restrict__ er,
                                  const float* __restrict__ ee, const int* __restrict__ src,
                                  const int* __restrict__ dst, float* __restrict__ lg,
                                  unsigned* __restrict__ mU) {
    int t = blockIdx.x * blockDim.x + threadIdx.x;
    if (t >= N_EDGES * 2) return;
    int e = t >> 1, h = t & 1;
    float l = el[src[e] * 2 + h] + er[dst[e] * 2 + h] + ee[t];
    l = (l > 0.f) ? l : 0.2f * l;
    lg[t] = l;
    atomicMax(&mU[dst[e] * 2 + h], f2ord(l));
}

// p = exp(logit - m[dst]); segment-sum into sB[dst]; lg overwritten with p
__global__ void p_sum_kernel(float* __restrict__ lg, const unsigned* __restrict__ mU,
                             const int* __restrict__ dst, float* __restrict__ sB) {
    int t = blockIdx.x * blockDim.x + threadIdx.x;
    if (t >= N_EDGES * 2) return;
    int e = t >> 1, h = t & 1;
    float m  = ord2f(mU[dst[e] * 2 + h]);
    float pv = expf(lg[t] - m);
    lg[t]    = pv;
    atomicAdd(&sB[dst[e] * 2 + h], pv);
}

// msg = (ft[src] + fe) * (p / s[dst])  scatter-added into agg[dst]; 4 cols / thread
__global__ void scatter_msg_kernel(const float* __restrict__ ft, const float* __restrict__ fe,
                                   const float* __restrict__ p, const float* __restrict__ sB,
                                   const int* __restrict__ src, const int* __restrict__ dst,
                                   float* __restrict__ agg) {
    int t = blockIdx.x * blockDim.x + threadIdx.x;
    if (t >= N_EDGES * 32) return;
    int e  = t >> 5;
    int c0 = (t & 31) << 2;          // column 0..124, groups of 4
    int h  = c0 >> 6;
    int sN = src[e], dN = dst[e];
    float alpha = p[e * 2 + h] / sB[dN * 2 + h];
    const float* fts = ft + (size_t)sN * 128 + c0;
    const float* fes = fe + (size_t)e * 128 + c0;
    float*       ag  = agg + (size_t)dN * 128 + c0;
#pragma unroll
    for (int i = 0; i < 4; ++i) atomicAdd(&ag[i], (fts[i] + fes[i]) * alpha);
}

// out[n,d] = mean over heads of (agg[n,h,d] + b[h,d])
__global__ void finalize_kernel(const float* __restrict__ agg, const float* __restrict__ bb,
                                float* __restrict__ xout) {
    int t = blockIdx.x * blockDim.x + threadIdx.x;
    if (t >= N_NODES * 64) return;
    int n = t >> 6, d = t & 63;
    float v0 = agg[(size_t)n * 128 + d] + bb[d];
    float v1 = agg[(size_t)n * 128 + 64 + d] + bb[64 + d];
    xout[t] = 0.5f * (v0 + v1);
}

// ---------------------------------------------------------------------------
// Host-side driver
// ---------------------------------------------------------------------------
static inline int cdiv(int a, int b) { return (a + b - 1) / b; }

struct WsPtrs {
    float* ft; float* fe; float* el; float* er; float* ee; float* lg;
    unsigned* mU; float* sB; float* agg; float* xb0; float* xb1;
};

static void run_edge_gat_layer(const float* xin, const float* eF, const int* src,
                               const int* dst, const float* Wn, const float* We,
                               const float* al, const float* ar, const float* ae,
                               const float* bb, float* xout, const WsPtrs& w,
                               hipStream_t stream) {
    // ft = xin @ Wn : [N,64]@[64,128]; 1250*8 tiles, 4 waves/block -> 2500 blocks
    gemm_wmma_f32<false><<<(N_NODES / 16) * (128 / 16) / 4, 128, 0, stream>>>(
        xin, Wn, w.ft, N_NODES, 64, 128, 64, 128, 128);
    // fe = eF @ We : [E,32]@[32,128]; 6250*8 tiles -> 12500 blocks
    gemm_wmma_f32<false><<<(N_EDGES / 16) * (128 / 16) / 4, 128, 0, stream>>>(
        eF, We, w.fe, N_EDGES, 32, 128, 32, 128, 128);

    node_attn_kernel<<<cdiv(N_NODES * 2, TPB), TPB, 0, stream>>>(w.ft, al, ar, w.el, w.er);
    edge_attn_kernel<<<cdiv(N_EDGES * 2, TPB), TPB, 0, stream>>>(w.fe, ae, w.ee);

    // zero m (uint 0 == below ord(-inf)), s, agg — laid out contiguously
    zero_kernel<<<cdiv(N_NODES * (2 + 2 + 128), TPB), TPB, 0, stream>>>(
        (float*)w.mU, N_NODES * (2 + 2 + 128));

    logits_max_kernel<<<cdiv(N_EDGES * 2, TPB), TPB, 0, stream>>>(w.el, w.er, w.ee, src, dst,
                                                                  w.lg, w.mU);
    p_sum_kernel<<<cdiv(N_EDGES * 2, TPB), TPB, 0, stream>>>(w.lg, w.mU, dst, w.sB);
    scatter_msg_kernel<<<cdiv(N_EDGES * 32, TPB), TPB, 0, stream>>>(w.ft, w.fe, w.lg, w.sB,
                                                                    src, dst, w.agg);
    finalize_kernel<<<cdiv(N_NODES * 64, TPB), TPB, 0, stream>>>(w.agg, bb, xout);
}

extern "C" void kernel_launch(void* const* d_in, const int* in_sizes, int n_in,
                              void* d_out, int out_size, void* d_ws, size_t ws_size,
                              hipStream_t stream) {
    // setup_inputs() dict order:
    // 0 x[S,N,64] 1 e[S,E,32] 2 src[S,E] 3 dst[S,E]
    // 4..9  enc0 {Wn,We,al,ar,ae,b}   10..15 enc1   16..21 dec0   22..27 dec1
    // 28 W_e2d[64,64]
    const float* x_all = (const float*)d_in[0];
    const float* e_all = (const float*)d_in[1];
    const int*   s_all = (const int*)d_in[2];
    const int*   d_all = (const int*)d_in[3];

    // Only the LAST snapshot (index 1) contributes to decoded[-1].
    const float* xS  = x_all + (size_t)1 * N_NODES * 64;
    const float* eS  = e_all + (size_t)1 * N_EDGES * 32;
    const int*   src = s_all + (size_t)1 * N_EDGES;
    const int*   dst = d_all + (size_t)1 * N_EDGES;

    const float* const* P = (const float* const*)d_in;  // convenient f32 view
    const float* W_e2d = P[28];

    // Workspace carve-up (floats). mU/sB/agg kept contiguous for one zero pass.
    float* ws = (float*)d_ws;
    size_t off = 0;
    WsPtrs w;
    w.ft  = ws + off; off += (size_t)N_NODES * 128;
    w.fe  = ws + off; off += (size_t)N_EDGES * 128;
    w.el  = ws + off; off += (size_t)N_NODES * 2;
    w.er  = ws + off; off += (size_t)N_NODES * 2;
    w.ee  = ws + off; off += (size_t)N_EDGES * 2;
    w.lg  = ws + off; off += (size_t)N_EDGES * 2;
    w.mU  = (unsigned*)(ws + off); off += (size_t)N_NODES * 2;
    w.sB  = ws + off; off += (size_t)N_NODES * 2;     // contiguous after mU
    w.agg = ws + off; off += (size_t)N_NODES * 128;   // contiguous after sB
    w.xb0 = ws + off; off += (size_t)N_NODES * 64;
    w.xb1 = ws + off; off += (size_t)N_NODES * 64;

    // encoder layer 0: x -> xb0
    run_edge_gat_layer(xS, eS, src, dst, P[4], P[5], P[6], P[7], P[8], P[9],
                       w.xb0, w, stream);
    // encoder layer 1: xb0 -> xb1
    run_edge_gat_layer(w.xb0, eS, src, dst, P[10], P[11], P[12], P[13], P[14], P[15],
                       w.xb1, w, stream);
    // encoder_to_decoder: xb0 = xb1 @ W_e2d.T  (B transposed GEMM, NC=64)
    gemm_wmma_f32<true><<<(N_NODES / 16) * (64 / 16) / 4, 128, 0, stream>>>(
        w.xb1, W_e2d, w.xb0, N_NODES, 64, 64, 64, 64, 64);
    // decoder layer 0: xb0 -> xb1
    run_edge_gat_layer(w.xb0, eS, src, dst, P[16], P[17], P[18], P[19], P[20], P[21],
                       w.xb1, w, stream);
    // decoder layer 1: xb1 -> d_out  [N,64] f32
    run_edge_gat_layer(w.xb1, eS, src, dst, P[22], P[23], P[24], P[25], P[26], P[27],
                       (float*)d_out, w, stream);
}